// FR_37958920962054
// MI455X (gfx1250) — compile-verified
//
#include <hip/hip_runtime.h>
#include <stdint.h>

// Problem constants (from reference setup_inputs): N=4, C=256, H=W=128, fp32.
#define NB 4
#define CC 256
#define HH 128
#define WW 128
#define HWSZ (HH * WW)
#define SCALE 0.125f

typedef int v4i_t __attribute__((ext_vector_type(4)));
typedef int v8i_t __attribute__((ext_vector_type(8)));

// ---------------------------------------------------------------------------
// Pass 1: NCHW -> NHWC transpose, staged through LDS by the Tensor Data Mover.
// One TDM descriptor per block moves a (C=256) x (Wtile=32) tile:
//   256 rows of 128B, row stride = H*W*4 = 64KB, into contiguous LDS with
//   1-DWORD padding every 32 DWORDs (row stride 33 DWORDs) so that the
//   transposed LDS reads (stride 132B across lanes) are bank-conflict-free.
// ---------------------------------------------------------------------------
__device__ __forceinline__ void tdm_load_tile_c_major(uint32_t lds_byte_off,
                                                      uint64_t gaddr_bytes) {
  // D# group 0 (CDNA5 ISA)
  v4i_t g0;
  g0[0] = 1;                                              // count=1, user mode
  g0[1] = (int)lds_byte_off;                              // lds_addr [63:32]
  g0[2] = (int)(uint32_t)(gaddr_bytes & 0xFFFFFFFFull);   // global_addr lo
  g0[3] = (int)(((uint32_t)(gaddr_bytes >> 32) & 0x01FFFFFFu) | (2u << 30)); // ga hi | type=2
  // D# group 1 (§8.4): data_size=4B (code 2), pad_enable, pad_interval=32dw
  // (code 4), pad_amount=1dw (code 0); tensor_dim0=32, tensor_dim1=256,
  // tile_dim0=32, tile_dim1=256, tile_dim2=0, dim0_stride=H*W=16384.
  v8i_t g1;
  g1[0] = (2 << 16) | (1 << 20) | (4 << 22);
  g1[1] = (int)((32u & 0xFFFFu) << 16);                   // tensor_dim0 lo16 in [31:16]
  g1[2] = (int)(((32u >> 16) & 0xFFFFu) | ((256u & 0xFFFFu) << 16)); // td0 hi | td1 lo
  g1[3] = (int)(((256u >> 16) & 0xFFFFu) | (32u << 16));  // td1 hi | tile_dim0=32
  g1[4] = (int)(256u);                                    // tile_dim1=256, tile_dim2=0
  g1[5] = 16384;                                          // dim0_stride lo32
  g1[6] = 0;                                              // dim0_stride hi | dim1_stride lo
  g1[7] = 0;
  v4i_t g2 = {0, 0, 0, 0};                                // 2D tile: groups 2/3 unused
  v4i_t g3 = {0, 0, 0, 0};
  asm volatile("tensor_load_to_lds %0, %1, %2, %3"
               :
               : "s"(g0), "s"(g1), "s"(g2), "s"(g3)
               : "memory");
}

__global__ __launch_bounds__(256) void FR_nchw_to_nhwc_tdm(
    const float* __restrict__ feat, float* __restrict__ nhwc) {
  __shared__ float tile[CC * 33 + 4];  // 33-dword padded rows (TDM pad feature)
  const int w0 = blockIdx.x * 32;
  const int h  = blockIdx.y;
  const int n  = blockIdx.z;

  if (threadIdx.x < 32) {  // wave 0 issues the DMA (EXEC ignored for TDM)
    const uint64_t g = (uint64_t)(uintptr_t)feat +
                       4ull * ((uint64_t)n * CC * HWSZ + (uint64_t)h * WW + (uint64_t)w0);
    // Generic shared pointer: low 32 bits are the LDS byte offset.
    tdm_load_tile_c_major((uint32_t)(uintptr_t)&tile[0], g);
#if defined(__has_builtin) && __has_builtin(__builtin_amdgcn_s_wait_tensorcnt)
    __builtin_amdgcn_s_wait_tensorcnt(0);
#else
    asm volatile("s_wait_tensorcnt 0x0" ::: "memory");
#endif
  }
  __syncthreads();

  // Write out transposed: thread t = channel c; stores coalesced over c.
  const int c = threadIdx.x;
  float* dst = nhwc + ((uint64_t)((n * HH + h) * WW + w0)) * CC + c;
#pragma unroll 8
  for (int w = 0; w < 32; ++w) {
    dst[(uint64_t)w * CC] = tile[c * 33 + w];  // stride 33 dwords: conflict-free
  }
}

// ---------------------------------------------------------------------------
// Pass 2: one wave32 per output pixel. Lane l owns channels {4l..4l+3} and
// {128+4l..131+4l} as two float4 accumulators. 20 corner gathers per pixel,
// each 2 coalesced b128 loads from the L2-resident NHWC copy.
// ---------------------------------------------------------------------------
__device__ __forceinline__ float4 fma4(float s, float4 a, float4 acc) {
  acc.x = fmaf(s, a.x, acc.x);
  acc.y = fmaf(s, a.y, acc.y);
  acc.z = fmaf(s, a.z, acc.z);
  acc.w = fmaf(s, a.w, acc.w);
  return acc;
}

__global__ __launch_bounds__(256) void FR_gather_refine(
    const float* __restrict__ nhwc, const float* __restrict__ bbox,
    float* __restrict__ out) {
  const int lane = threadIdx.x & 31;
  const int pix  = blockIdx.x * 8 + (threadIdx.x >> 5);
  const int n  = pix >> 14;      // / (H*W)
  const int hw = pix & (HWSZ - 1);
  const int h  = hw >> 7;        // / W
  const int w  = hw & (WW - 1);

  // Box params (uniform within the wave; hardware broadcasts the load).
  const float* bb = bbox + (size_t)pix * 5;
  const float cx = bb[0], cy = bb[1], bwid = bb[2], bhei = bb[3], th = bb[4];
  const float ct = cosf(th), st = sinf(th);
  const float vx = bwid * ct * 0.5f, vy = bwid * st * 0.5f;
  const float wxr = -bhei * st * 0.5f, wyr = bhei * ct * 0.5f;
  const float pxs[5] = {cx, cx + vx + wxr, cx - vx + wxr, cx - vx - wxr, cx + vx - wxr};
  const float pys[5] = {cy, cy + vy + wyr, cy - vy + wyr, cy - vy - wyr, cy + vy - wyr};

  float4 acc0 = {0.f, 0.f, 0.f, 0.f};
  float4 acc1 = {0.f, 0.f, 0.f, 0.f};
  const size_t base_n = (size_t)n * HH * WW;

#pragma unroll
  for (int p = 0; p < 5; ++p) {
    float x = pxs[p] * SCALE;
    float y = pys[p] * SCALE;
    const bool valid = (y >= -1.0f) && (y <= (float)HH) && (x >= -1.0f) && (x <= (float)WW);
    y = fmaxf(y, 0.0f);
    x = fmaxf(x, 0.0f);
    int y_low = min((int)floorf(y), HH - 1);
    int x_low = min((int)floorf(x), WW - 1);
    if (y_low >= HH - 1) y = (float)y_low;  // mmcv border rule: ly -> 0
    if (x_low >= WW - 1) x = (float)x_low;
    const int y_high = min(y_low + 1, HH - 1);
    const int x_high = min(x_low + 1, WW - 1);
    const float ly = y - (float)y_low, lx = x - (float)x_low;
    const float hy = 1.0f - ly, hx = 1.0f - lx;
    float w00 = hy * hx, w01 = hy * lx, w10 = ly * hx, w11 = ly * lx;
    if (!valid) { w00 = w01 = w10 = w11 = 0.0f; }

    const float4* c00 = (const float4*)(nhwc + (base_n + (size_t)y_low  * WW + x_low ) * CC);
    const float4* c01 = (const float4*)(nhwc + (base_n + (size_t)y_low  * WW + x_high) * CC);
    const float4* c10 = (const float4*)(nhwc + (base_n + (size_t)y_high * WW + x_low ) * CC);
    const float4* c11 = (const float4*)(nhwc + (base_n + (size_t)y_high * WW + x_high) * CC);

    float4 a, b;
    a = c00[lane]; b = c00[lane + 32];
    acc0 = fma4(w00, a, acc0); acc1 = fma4(w00, b, acc1);
    a = c01[lane]; b = c01[lane + 32];
    acc0 = fma4(w01, a, acc0); acc1 = fma4(w01, b, acc1);
    a = c10[lane]; b = c10[lane + 32];
    acc0 = fma4(w10, a, acc0); acc1 = fma4(w10, b, acc1);
    a = c11[lane]; b = c11[lane + 32];
    acc0 = fma4(w11, a, acc0); acc1 = fma4(w11, b, acc1);
  }

  // Residual add from the NHWC copy (contiguous), then scatter to NCHW output.
  const float4* self = (const float4*)(nhwc + (size_t)pix * CC);
  const float4 r0 = self[lane];
  const float4 r1 = self[lane + 32];

  float* o = out + (size_t)n * CC * HWSZ + (size_t)h * WW + w;
  const int c0 = lane * 4;
  o[(size_t)(c0 + 0) * HWSZ] = acc0.x + r0.x;
  o[(size_t)(c0 + 1) * HWSZ] = acc0.y + r0.y;
  o[(size_t)(c0 + 2) * HWSZ] = acc0.z + r0.z;
  o[(size_t)(c0 + 3) * HWSZ] = acc0.w + r0.w;
  const int c1 = 128 + lane * 4;
  o[(size_t)(c1 + 0) * HWSZ] = acc1.x + r1.x;
  o[(size_t)(c1 + 1) * HWSZ] = acc1.y + r1.y;
  o[(size_t)(c1 + 2) * HWSZ] = acc1.z + r1.z;
  o[(size_t)(c1 + 3) * HWSZ] = acc1.w + r1.w;
}

extern "C" void kernel_launch(void* const* d_in, const int* in_sizes, int n_in,
                              void* d_out, int out_size, void* d_ws, size_t ws_size,
                              hipStream_t stream) {
  (void)in_sizes; (void)n_in; (void)out_size; (void)ws_size;
  const float* feat = (const float*)d_in[0];   // [N,C,H,W] fp32
  const float* bbox = (const float*)d_in[1];   // [N,H,W,5] fp32
  float* out  = (float*)d_out;                 // [N,C,H,W] fp32
  float* nhwc = (float*)d_ws;                  // needs N*H*W*C*4 = 64 MiB scratch

  // Pass 1: TDM-staged NCHW -> NHWC transpose (L2-resident working set).
  dim3 g1(WW / 32, HH, NB);
  FR_nchw_to_nhwc_tdm<<<g1, 256, 0, stream>>>(feat, nhwc);

  // Pass 2: one wave per pixel; 5-point bilinear gather + residual + scatter.
  const int nPix = NB * HH * WW;               // 65536 pixels = 65536 waves
  FR_gather_refine<<<nPix / 8, 256, 0, stream>>>(nhwc, bbox, out);
}